// BGNetwork_37151467110536
// MI455X (gfx1250) — compile-verified
//
#include <hip/hip_runtime.h>

// ---------------------------------------------------------------------------
// BGNetwork on MI455X (gfx1250, wave32).
//   Outputs (13 floats): [-v_gpi(4), t(1), dp_output(4), ip_output(4)]
// Strategy:
//   K1: D1 pathway via V_WMMA_F32_16X16X4_F32 (confirmed lowering)
//   K2: chunk affine summaries; one chunk per lane (float4 = 4 arms);
//       double-buffered GLOBAL_LOAD_ASYNC_TO_LDS_B128 staging (ASYNCcnt),
//       wave-uniform tile counts so the issue/wait contract is exact
//   K3: 1-wave affine scan over chunk summaries via shfl_up composition
//   K4: per-chunk replay from true boundary state (unroll-8 b128 loads),
//       in-lane first-crossing detection
//   K5: 1-wave finalize -> d_out
// ---------------------------------------------------------------------------

#define NUM_ARMS  4
#define ALPHA     0.1f          // DT / TAU_GPI = 0.01 / 0.1
#define CDECAY    0.9f          // 1 - ALPHA
#define GPI_THR   10.0f
#define K_INT_MAX 0x7fffffff
#define TILE      8             // steps staged per async tile (128B per lane)

typedef __attribute__((ext_vector_type(2))) float v2f;
typedef __attribute__((ext_vector_type(8))) float v8f;

#if defined(__has_builtin)
#if __has_builtin(__builtin_amdgcn_s_wait_asynccnt)
#define WAIT_ASYNC(n) __builtin_amdgcn_s_wait_asynccnt(n)
#endif
#endif
#ifndef WAIT_ASYNC
#define WAIT_ASYNC(n) asm volatile("s_wait_asynccnt %0" ::"i"(n) : "memory")
#endif

__device__ __forceinline__ float sigmoidf(float x) {
    // v_exp_f32 + v_rcp_f32 (avoid the IEEE div_scale/div_fixup chain)
    return __builtin_amdgcn_rcpf(1.0f + __expf(-x));
}

__device__ __forceinline__ float round2f(float x) {
    // torch.round(x, decimals=2): RNE == rintf with default round mode
    return rintf(x * 100.0f) / 100.0f;
}

// Issue one per-lane 16B async copy: global -> this lane's LDS slot.
__device__ __forceinline__ void async_copy16(const void* lds_dst, const void* gsrc) {
    unsigned loff = (unsigned)(unsigned long long)(uintptr_t)lds_dst; // LDS byte offset
    unsigned long long ga = (unsigned long long)(uintptr_t)gsrc;
    asm volatile("global_load_async_to_lds_b128 %0, %1, off"
                 :: "v"(loff), "v"(ga)
                 : "memory");
}

// One 1x4 row (r0..r3 broadcast) times 4x4 W^T inside a 16x16x4 f32 WMMA tile.
// Returns D[0][n], valid on lanes 0..15 (n = lane).
__device__ __forceinline__ float tiny_matvec_wmma(float r0, float r1, float r2,
                                                  float r3, const float* W,
                                                  int lane) {
    const int n  = lane & 15;
    const int hi = lane >> 4;
    // A 16x4 f32: lane<16 holds A[M][K0..K1]; lane>=16 holds A[M][K2..K3]
    v2f a; a.x = 0.0f; a.y = 0.0f;
    if (lane == 0)       { a.x = r0; a.y = r1; }
    else if (lane == 16) { a.x = r2; a.y = r3; }
    // B 4x16 f32: VGPR0 lanes0-15 K=0 / lanes16-31 K=2; VGPR1 K=1 / K=3
    v2f b; b.x = 0.0f; b.y = 0.0f;
    if (n < NUM_ARMS) {
        b.x = W[n * 4 + (hi ? 2 : 0)];   // B[k][n] = W[n][k]  (x @ W^T)
        b.y = W[n * 4 + (hi ? 3 : 1)];
    }
    v8f c = {0.f, 0.f, 0.f, 0.f, 0.f, 0.f, 0.f, 0.f};
    v8f d = __builtin_amdgcn_wmma_f32_16x16x4_f32(false, a, false, b,
                                                  (short)0, c, false, false);
    return d[0];
}

// --------------------------- K1: D1 pathway --------------------------------
__global__ __launch_bounds__(32) void bg_d1_pathway(
    const float* __restrict__ w1, const float* __restrict__ b1,
    const float* __restrict__ w2, const float* __restrict__ b2,
    float* __restrict__ drive, float* __restrict__ dp) {
    const int lane = threadIdx.x;
    const int n    = lane & 15;
    const int hi   = lane >> 4;

    float pre1 = tiny_matvec_wmma(1.0f, 1.0f, 1.0f, 1.0f, w1, lane);
    float h = 0.0f;
    if (hi == 0 && n < NUM_ARMS) h = sigmoidf(pre1 + b1[n]);
    float h0 = __shfl(h, 0, 32), h1 = __shfl(h, 1, 32);
    float h2 = __shfl(h, 2, 32), h3 = __shfl(h, 3, 32);

    float pre2 = tiny_matvec_wmma(h0, h1, h2, h3, w2, lane);
    if (hi == 0 && n < NUM_ARMS) {
        float o  = sigmoidf(pre2 + b2[n]);
        drive[n] = -0.5f * o;   // -D1_AMP * D1_output
        dp[n]    =  0.5f * o;   //  D1_AMP * D1_output
    }
}

// ----------------- K2: chunk affine summaries (one chunk / lane) -----------
// Summary: v_out = Apow * v_in + Bsum  with per-step  v' = c*v + alpha*(drive+ip)
// Issue/wait contract: every tile issues exactly TILE async b128 copies
// (addresses clamped, never skipped), so ASYNCcnt arithmetic is exact:
//   before processing tile k (next tile issued): cnt = 2*TILE -> wait<=TILE
//   before processing the last tile:             wait<=0
__global__ __launch_bounds__(32) void bg_pass1_summaries(
    const float* __restrict__ stn, const float* __restrict__ drive,
    float* __restrict__ Bsum, float* __restrict__ Apow,
    int T, int N, int L) {
    __shared__ float4 stage[2][32][TILE + 1];   // +1 pad: avoid 16-way bank conflicts
    const int lane  = threadIdx.x;
    const int chunk = blockIdx.x * 32 + lane;
    const float4* stn4 = (const float4*)stn;

    const int s0   = chunk * L;
    const int s1   = min(T, s0 + L);
    const int smax = T - 1;
    const int ntiles = (L + TILE - 1) / TILE;   // wave-uniform

    float4 drv = *(const float4*)drive;
    float4 B = {0.f, 0.f, 0.f, 0.f};
    float  A = 1.0f;

    auto issue = [&](int tile) {
        const int buf = tile & 1;
#pragma unroll
        for (int j = 0; j < TILE; ++j) {
            int s  = s0 + tile * TILE + j;
            int sc = (s < smax) ? s : smax;     // clamp: always issue TILE copies
            async_copy16(&stage[buf][lane][j], stn4 + sc);
        }
    };

    issue(0);
    for (int tile = 0; tile < ntiles; ++tile) {
        if (tile + 1 < ntiles) { issue(tile + 1); WAIT_ASYNC(TILE); }
        else                   { WAIT_ASYNC(0); }
        const int buf = tile & 1;
#pragma unroll
        for (int j = 0; j < TILE; ++j) {
            int s = s0 + tile * TILE + j;
            if (s < s1) {
                float4 x = stage[buf][lane][j];
                B.x = CDECAY * B.x + ALPHA * (drv.x + round2f(x.x));
                B.y = CDECAY * B.y + ALPHA * (drv.y + round2f(x.y));
                B.z = CDECAY * B.z + ALPHA * (drv.z + round2f(x.z));
                B.w = CDECAY * B.w + ALPHA * (drv.w + round2f(x.w));
                A *= CDECAY;
            }
        }
    }
    ((float4*)Bsum)[chunk] = B;
    Apow[chunk] = A;
}

// -------- K3: 1-wave parallel scan of affine maps over chunks --------------
// Composition of v->A*v+B maps; 5 shfl_up rounds per 32 chunks.
__global__ __launch_bounds__(32) void bg_scan_chunks(
    const float* __restrict__ Bsum, const float* __restrict__ Apow,
    float* __restrict__ vb, int N) {
    const int lane = threadIdx.x;
    float4 carry = {0.f, 0.f, 0.f, 0.f};   // state entering current group

    for (int base = 0; base < N; base += 32) {
        const int g = base + lane;
        float  A = Apow[g];
        float4 B = ((const float4*)Bsum)[g];
        // inclusive Hillis-Steele scan of affine maps
#pragma unroll
        for (int d = 1; d < 32; d <<= 1) {
            float  Ap = __shfl_up(A, d, 32);
            float4 Bp;
            Bp.x = __shfl_up(B.x, d, 32); Bp.y = __shfl_up(B.y, d, 32);
            Bp.z = __shfl_up(B.z, d, 32); Bp.w = __shfl_up(B.w, d, 32);
            if (lane >= d) {
                B.x += A * Bp.x; B.y += A * Bp.y;
                B.z += A * Bp.z; B.w += A * Bp.w;
                A *= Ap;
            }
        }
        // exclusive prefix (lane-1 inclusive) applied to group-entry carry
        float  Ae = __shfl_up(A, 1, 32);
        float4 Be;
        Be.x = __shfl_up(B.x, 1, 32); Be.y = __shfl_up(B.y, 1, 32);
        Be.z = __shfl_up(B.z, 1, 32); Be.w = __shfl_up(B.w, 1, 32);
        float4 vstart;
        if (lane == 0) vstart = carry;
        else {
            vstart.x = Ae * carry.x + Be.x; vstart.y = Ae * carry.y + Be.y;
            vstart.z = Ae * carry.z + Be.z; vstart.w = Ae * carry.w + Be.w;
        }
        ((float4*)vb)[g] = vstart;          // state entering chunk g
        // group total (lane 31 inclusive) advances the carry
        float  At = __shfl(A, 31, 32);
        float4 Bt;
        Bt.x = __shfl(B.x, 31, 32); Bt.y = __shfl(B.y, 31, 32);
        Bt.z = __shfl(B.z, 31, 32); Bt.w = __shfl(B.w, 31, 32);
        carry.x = At * carry.x + Bt.x; carry.y = At * carry.y + Bt.y;
        carry.z = At * carry.z + Bt.z; carry.w = At * carry.w + Bt.w;
    }
}

// ------- K4: exact replay from true boundary state + crossing detect -------
__global__ __launch_bounds__(32) void bg_pass2_replay(
    const float* __restrict__ stn, const float* __restrict__ drive,
    const float* __restrict__ vb, int* __restrict__ cstep,
    float* __restrict__ vcross, float* __restrict__ vend,
    int T, int N, int L) {
    const int chunk = blockIdx.x * 32 + threadIdx.x;
    const float4* stn4 = (const float4*)stn;

    const int s0 = chunk * L;
    const int s1 = min(T, s0 + L);

    float4 drv = *(const float4*)drive;
    float4 v   = ((const float4*)vb)[chunk];
    int    cross = K_INT_MAX;
    float4 vc = {0.f, 0.f, 0.f, 0.f};

#pragma unroll 8
    for (int s = s0; s < s1; ++s) {
        __builtin_prefetch(stn4 + s + 32, 0, 1);   // global_prefetch_b8
        float4 x = stn4[s];
        // matches reference op order: v + alpha*(drive + ip - v)
        v.x = v.x + ALPHA * (drv.x + round2f(x.x) - v.x);
        v.y = v.y + ALPHA * (drv.y + round2f(x.y) - v.y);
        v.z = v.z + ALPHA * (drv.z + round2f(x.z) - v.z);
        v.w = v.w + ALPHA * (drv.w + round2f(x.w) - v.w);
        float m = fmaxf(fmaxf(-v.x, -v.y), fmaxf(-v.z, -v.w));
        if (m > GPI_THR && cross == K_INT_MAX) { cross = s; vc = v; }
    }
    ((float4*)vend)[chunk]   = v;
    ((float4*)vcross)[chunk] = vc;
    cstep[chunk] = cross;
}

// ------------------------- K5: finalize outputs ----------------------------
__global__ __launch_bounds__(32) void bg_finalize(
    const float* __restrict__ stn, const int* __restrict__ cstep,
    const float* __restrict__ vcross, const float* __restrict__ vend,
    const float* __restrict__ dp, float* __restrict__ out, int T, int N) {
    const int lane = threadIdx.x;
    int best = K_INT_MAX, bestg = 0;
    for (int g = lane; g < N; g += 32) {
        int cs = cstep[g];
        if (cs < best) { best = cs; bestg = g; }
    }
    for (int off = 16; off > 0; off >>= 1) {
        int ob = __shfl_xor(best, off, 32);
        int og = __shfl_xor(bestg, off, 32);
        if (ob < best) { best = ob; bestg = og; }
    }
    if (lane < NUM_ARMS) {
        float vfin, d2;
        int t;
        if (best != K_INT_MAX) {                   // GPi fired at step `best`
            t    = best + 1;
            vfin = vcross[bestg * 4 + lane];
            d2   = stn[best * 4 + lane];
        } else {                                   // ran all T steps
            t    = T;
            vfin = vend[(N - 1) * 4 + lane];
            d2   = stn[(T - 1) * 4 + lane];
        }
        out[lane]     = -vfin;                     // v_gpi_out
        out[5 + lane] = dp[lane];                  // dp_output
        out[9 + lane] = round2f(d2);               // ip_output
        if (lane == 0) out[4] = (float)t;          // t (int32 in ref; stored as f32)
    }
}

// ---------------------------------------------------------------------------
extern "C" void kernel_launch(void* const* d_in, const int* in_sizes, int n_in,
                              void* d_out, int out_size, void* d_ws, size_t ws_size,
                              hipStream_t stream) {
    const float* stn = (const float*)d_in[0];   // (1, T, 4)
    const float* w1  = (const float*)d_in[1];   // str_d1_w (4,4)
    const float* b1  = (const float*)d_in[2];   // str_d1_b (4,)
    const float* w2  = (const float*)d_in[3];   // d1_gpi_w (4,4)
    const float* b2  = (const float*)d_in[4];   // d1_gpi_b (4,)
    // d_in[5] snc_w, d_in[6] snc_b: feed `value`, not an output.

    const int T = in_sizes[0] / NUM_ARMS;

    // Chunk count: multiple of 32, fits workspace (32 + 72*N bytes).
    int N = 1024;
    while (N > 32 && (size_t)(32 + 72 * (size_t)N) > ws_size) N >>= 1;
    const int L = (T + N - 1) / N;

    float* ws     = (float*)d_ws;
    float* drive  = ws;                        // 4
    float* dp     = ws + 4;                    // 4
    float* Bsum   = ws + 8;                    // N*4   (16B aligned)
    float* Apow   = Bsum + 4 * N;              // N
    float* vb     = Apow + N;                  // N*4   (16B aligned)
    int*   cstep  = (int*)(vb + 4 * N);        // N
    float* vcross = (float*)(cstep + N);       // N*4   (16B aligned)
    float* vend   = vcross + 4 * N;            // N*4   (16B aligned)
    float* out    = (float*)d_out;

    const int blocks = N / 32;

    hipLaunchKernelGGL(bg_d1_pathway, dim3(1), dim3(32), 0, stream,
                       w1, b1, w2, b2, drive, dp);
    hipLaunchKernelGGL(bg_pass1_summaries, dim3(blocks), dim3(32), 0, stream,
                       stn, drive, Bsum, Apow, T, N, L);
    hipLaunchKernelGGL(bg_scan_chunks, dim3(1), dim3(32), 0, stream,
                       Bsum, Apow, vb, N);
    hipLaunchKernelGGL(bg_pass2_replay, dim3(blocks), dim3(32), 0, stream,
                       stn, drive, vb, cstep, vcross, vend, T, N, L);
    hipLaunchKernelGGL(bg_finalize, dim3(1), dim3(32), 0, stream,
                       stn, cstep, vcross, vend, dp, out, T, N);
}